// FCAGAT_1348619730981
// MI455X (gfx1250) — compile-verified
//
#include <hip/hip_runtime.h>
#include <math.h>

// ---------------- constants from the reference ----------------
#define HEADS 2
#define CDIM  64
#define HC    128      // H*C
#define FUSEDD 384     // 3 * HC
#define HID   128      // classifier hidden

typedef __attribute__((ext_vector_type(2))) float v2f;
typedef __attribute__((ext_vector_type(8))) float v8f;
typedef __attribute__((ext_vector_type(4))) unsigned int u32x4;
typedef __attribute__((ext_vector_type(8))) unsigned int u32x8;

// -------- fp32 WMMA tile: D(16x16) += A(16x4) * B(4x16) --------
// A layout: lanes 0-15 hold M=lane, VGPR0=K0 VGPR1=K1; lanes 16-31 hold K2/K3.
// B layout: row-striped: VGPR0 lanes0-15 => K0,N=lane; lanes16-31 => K2. VGPR1 => K1/K3.
// C/D: VGPR r: lanes0-15 row M=r, lanes16-31 row M=r+8, N=lane&15.
__device__ __forceinline__ v8f wmma_f32(v2f a, v2f b, v8f c) {
    return __builtin_amdgcn_wmma_f32_16x16x4_f32(
        /*neg_a=*/false, a, /*neg_b=*/false, b,
        /*c_mod=*/(short)0, c, /*reuse_a=*/false, /*reuse_b=*/false);
}

// ============================================================
// TDM: stage a rows x 128 f32 panel (row-major, stride 128) from
// global into the LDS buffer `lds_dst` via tensor_load_to_lds.
// D# built per CDNA5 ISA ch.8 (group0 + group1, 2-D tile).
// The LDS pointer is passed into the asm (operand %2, unused in
// the template) so the buffer formally escapes and the "memory"
// clobber forces post-DMA reloads; its low 32 bits are the LDS
// byte offset (generic LDS addr = {SHARED_BASE, offset[31:0]}).
// ============================================================
__device__ __forceinline__ void tdm_stage_panel(const float* gsrc, float* lds_dst,
                                                int rows, int cols) {
    unsigned long long ga = (unsigned long long)(size_t)gsrc;
    unsigned lds_off = (unsigned)(size_t)lds_dst;   // LDS byte offset
    u32x4 g0;
    g0.x = 0x1u;                                    // count=1, no gather/iterate
    g0.y = lds_off;                                 // lds_addr (bytes)
    g0.z = (unsigned)ga;                            // global_addr[31:0]
    g0.w = (unsigned)((ga >> 32) & 0x1FFFFFFu)      // global_addr[56:32]
         | (2u << 30);                              // type=2 ("image")
    u32x8 g1;
    g1.s0 = 2u << 16;                               // data_size=4B (code 2)
    g1.s1 = ((unsigned)cols & 0xFFFFu) << 16;       // tensor_dim0[15:0] @bits63:48
    g1.s2 = (((unsigned)cols >> 16) & 0xFFFFu)      // tensor_dim0[31:16]
          | (((unsigned)rows & 0xFFFFu) << 16);     // tensor_dim1[15:0]
    g1.s3 = (((unsigned)rows >> 16) & 0xFFFFu)      // tensor_dim1[31:16]
          | (((unsigned)cols & 0xFFFFu) << 16);     // tile_dim0 @bits127:112
    g1.s4 = (unsigned)rows;                         // tile_dim1 (tile_dim2=0)
    g1.s5 = (unsigned)cols;                         // tensor_dim0_stride[31:0]
    g1.s6 = 0u;                                     // stride hi / dim1_stride lo
    g1.s7 = 0u;
    asm volatile("tensor_load_to_lds %0, %1"
                 :: "s"(g0), "s"(g1), "r"(lds_dst) : "memory");
    __builtin_amdgcn_s_wait_tensorcnt(0);
}

// ============================================================
// XW = X[n,din] @ W[din,128]; 8 waves/block, each wave owns a
// 16x128 tile; W panel staged once into LDS by the TDM (wave 0).
// ============================================================
__global__ void __launch_bounds__(256)
gat_gemm_xw(const float* __restrict__ X, const float* __restrict__ W,
            float* __restrict__ XW, int n, int din) {
    __shared__ float WL[128 * HC];         // 64KB weight panel
    const int tid  = threadIdx.x;
    const int wv   = tid >> 5;             // wave 0..7
    const int lane = tid & 31;
    const int half = lane >> 4;
    const int l16  = lane & 15;
    const int row0 = (blockIdx.x * 8 + wv) * 16;

    if (tid < 32) tdm_stage_panel(W, WL, din, HC);   // wave 0 issues DMA + waits
    __syncthreads();

    v8f acc[8];
#pragma unroll
    for (int t = 0; t < 8; ++t)
#pragma unroll
        for (int r = 0; r < 8; ++r) acc[t][r] = 0.0f;

    int ar = row0 + l16; if (ar >= n) ar = n - 1;    // clamp loads
    const float* __restrict__ xrow = X + (size_t)ar * din;

    for (int k0 = 0; k0 < din; k0 += 4) {
        const int ak = k0 + 2 * half;
        v2f a;
        a.x = xrow[ak];
        a.y = xrow[ak + 1];
        const float* w0 = WL + ak * HC;
#pragma unroll
        for (int t = 0; t < 8; ++t) {
            const int nc = t * 16 + l16;
            v2f b;
            b.x = w0[nc];
            b.y = w0[HC + nc];
            acc[t] = wmma_f32(a, b, acc[t]);
        }
    }
#pragma unroll
    for (int t = 0; t < 8; ++t) {
        const int col = t * 16 + l16;
#pragma unroll
        for (int r = 0; r < 8; ++r) {
            const int row = row0 + r + 8 * half;
            if (row < n) XW[(size_t)row * HC + col] = acc[t][r];
        }
    }
}

// ============================================================
// a_src[n,H], a_dst[n,H]: per-(node,head) dot over C=64
// ============================================================
__global__ void gat_att_scores(const float* __restrict__ XW,
                               const float* __restrict__ att_s,
                               const float* __restrict__ att_d,
                               float* __restrict__ a_src, float* __restrict__ a_dst,
                               int n) {
    int idx = blockIdx.x * blockDim.x + threadIdx.x;
    if (idx >= n * HEADS) return;
    const int node = idx / HEADS, h = idx % HEADS;
    const float* __restrict__ row = XW + (size_t)node * HC + h * CDIM;
    const float* __restrict__ as  = att_s + h * CDIM;
    const float* __restrict__ ad  = att_d + h * CDIM;
    float ss = 0.f, sd = 0.f;
#pragma unroll 4
    for (int c = 0; c < CDIM; ++c) { float v = row[c]; ss += v * as[c]; sd += v * ad[c]; }
    a_src[idx] = ss;
    a_dst[idx] = sd;
}

// ============================================================
// init m=-inf, s=0 (n*H)  and agg=0 (n*HC)
// ============================================================
__global__ void gat_init(float* __restrict__ m, float* __restrict__ s,
                         float* __restrict__ agg, int n) {
    int i = blockIdx.x * blockDim.x + threadIdx.x;
    if (i < n * HEADS) { m[i] = -INFINITY; s[i] = 0.f; }
    if (i < n * HC)    agg[i] = 0.f;
}

// monotone float atomic-max without relying on float atomicMax availability
__device__ __forceinline__ void atomicMaxF32(float* addr, float val) {
    if (val >= 0.f) atomicMax((int*)addr, __float_as_int(val));
    else            atomicMin((unsigned int*)addr, __float_as_uint(val));
}

__device__ __forceinline__ float leaky02(float x) { return x > 0.f ? x : 0.2f * x; }

// ============================================================
// pass 1: segment max of leaky_relu(a_src[src]+a_dst[dst]) per dst
// ============================================================
__global__ void gat_edge_max(const int* __restrict__ ei, int E,
                             const float* __restrict__ a_src,
                             const float* __restrict__ a_dst,
                             float* __restrict__ m, int n) {
    int e = blockIdx.x * blockDim.x + threadIdx.x;
    const int total = E + n;                    // +N self loops
    if (e >= total) return;
    const int src = (e < E) ? ei[e]     : (e - E);
    const int dst = (e < E) ? ei[E + e] : (e - E);
#pragma unroll
    for (int h = 0; h < HEADS; ++h) {
        float v = leaky02(a_src[src * HEADS + h] + a_dst[dst * HEADS + h]);
        atomicMaxF32(&m[dst * HEADS + h], v);
    }
}

// ============================================================
// pass 2: s[dst] += exp(e - m[dst])
// ============================================================
__global__ void gat_edge_sum(const int* __restrict__ ei, int E,
                             const float* __restrict__ a_src,
                             const float* __restrict__ a_dst,
                             const float* __restrict__ m,
                             float* __restrict__ s, int n) {
    int e = blockIdx.x * blockDim.x + threadIdx.x;
    const int total = E + n;
    if (e >= total) return;
    const int src = (e < E) ? ei[e]     : (e - E);
    const int dst = (e < E) ? ei[E + e] : (e - E);
#pragma unroll
    for (int h = 0; h < HEADS; ++h) {
        float v  = leaky02(a_src[src * HEADS + h] + a_dst[dst * HEADS + h]);
        float ex = __expf(v - m[dst * HEADS + h]);
        atomicAdd(&s[dst * HEADS + h], ex);
    }
}

// ============================================================
// pass 3: agg[dst,:] += XW[src,:] * alpha   (one wave per edge,
// each lane handles 4 contiguous features; 4*32 = 128 = HC)
// ============================================================
__global__ void gat_edge_aggregate(const int* __restrict__ ei, int E,
                                   const float* __restrict__ a_src,
                                   const float* __restrict__ a_dst,
                                   const float* __restrict__ m,
                                   const float* __restrict__ s,
                                   const float* __restrict__ XW,
                                   float* __restrict__ agg, int n) {
    const long gid = (long)blockIdx.x * blockDim.x + threadIdx.x;
    const int e    = (int)(gid >> 5);
    const int lane = (int)(gid & 31);
    const int total = E + n;
    if (e >= total) return;
    const int src = (e < E) ? ei[e]     : (e - E);
    const int dst = (e < E) ? ei[E + e] : (e - E);

    // pre-touch the gathered feature row (global_prefetch_b8)
    __builtin_prefetch(XW + (size_t)src * HC + lane * 4, 0, 3);

    const int h = (lane * 4) / CDIM;            // all 4 features in same head
    float v  = leaky02(a_src[src * HEADS + h] + a_dst[dst * HEADS + h]);
    float ex = __expf(v - m[dst * HEADS + h]);
    float alpha = ex / s[dst * HEADS + h];

    const float4 xv = *(const float4*)(XW + (size_t)src * HC + lane * 4);
    float* out = agg + (size_t)dst * HC + lane * 4;
    atomicAdd(out + 0, xv.x * alpha);
    atomicAdd(out + 1, xv.y * alpha);
    atomicAdd(out + 2, xv.z * alpha);
    atomicAdd(out + 3, xv.w * alpha);
}

// ============================================================
// ELU(agg + bias) -> Hcat slice [n, 384] at column cat*128
// ============================================================
__global__ void gat_elu_concat(const float* __restrict__ agg,
                               const float* __restrict__ bias,
                               float* __restrict__ Hcat, int n, int cat) {
    int i = blockIdx.x * blockDim.x + threadIdx.x;
    if (i >= n * HC) return;
    const int node = i / HC, f = i % HC;
    float v = agg[i] + bias[f];
    v = v > 0.f ? v : (__expf(v) - 1.f);
    Hcat[(size_t)node * FUSEDD + cat * HC + f] = v;
}

// ============================================================
// classifier: out = relu(Hcat @ Wc1 + bc1) @ Wc2 + bc2
// 8 waves/block (16 rows each). Wc1 staged in LDS in 3 chunks of
// 128 K-rows (64KB). ReLU tile -> LDS, per-row dot with Wc2.
// ============================================================
__global__ void __launch_bounds__(256)
fused_classifier(const float* __restrict__ Hcat,
                 const float* __restrict__ Wc1, const float* __restrict__ bc1,
                 const float* __restrict__ Wc2, const float* __restrict__ bc2,
                 float* __restrict__ out, int n) {
    __shared__ float WL[128 * HID];         // 64KB: one K-chunk of Wc1
    __shared__ float hid[8][16 * HID];      // 64KB: per-wave relu'd tile
    const int tid  = threadIdx.x;
    const int wv   = tid >> 5;
    const int lane = tid & 31;
    const int half = lane >> 4;
    const int l16  = lane & 15;
    const int row0 = (blockIdx.x * 8 + wv) * 16;

    v8f acc[8];
#pragma unroll
    for (int t = 0; t < 8; ++t)
#pragma unroll
        for (int r = 0; r < 8; ++r) acc[t][r] = 0.0f;

    int ar = row0 + l16; if (ar >= n) ar = n - 1;
    const float* __restrict__ hrow = Hcat + (size_t)ar * FUSEDD;

    for (int chunk = 0; chunk < 3; ++chunk) {
        __syncthreads();                    // previous chunk fully consumed
        // cooperative stage of Wc1 rows [chunk*128, chunk*128+128)
        const float4* src4 = (const float4*)(Wc1 + (size_t)chunk * 128 * HID);
        float4*       dst4 = (float4*)WL;
        for (int i = tid; i < 128 * HID / 4; i += 256) dst4[i] = src4[i];
        __syncthreads();

        const int kbase = chunk * 128;
        for (int k0 = 0; k0 < 128; k0 += 4) {
            const int ak = k0 + 2 * half;
            v2f a;
            a.x = hrow[kbase + ak];
            a.y = hrow[kbase + ak + 1];
            const float* w0 = WL + ak * HID;
#pragma unroll
            for (int t = 0; t < 8; ++t) {
                const int nc = t * 16 + l16;
                v2f b;
                b.x = w0[nc];
                b.y = w0[HID + nc];
                acc[t] = wmma_f32(a, b, acc[t]);
            }
        }
    }
    // bias + relu into this wave's LDS tile [16 x 128]
#pragma unroll
    for (int t = 0; t < 8; ++t) {
        const int col = t * 16 + l16;
        const float bb = bc1[col];
#pragma unroll
        for (int r = 0; r < 8; ++r) {
            const int row = r + 8 * half;
            float v = acc[t][r] + bb;
            hid[wv][row * HID + col] = v > 0.f ? v : 0.f;
        }
    }
    __syncthreads();                        // covers the DS RAW hazard on hid[]
    // per-row dot with Wc2: lanes 0-15 cols [0,64), lanes 16-31 cols [64,128)
    float sum = 0.f;
    const int c0 = half * 64;
#pragma unroll 4
    for (int c = 0; c < 64; ++c) sum += hid[wv][l16 * HID + c0 + c] * Wc2[c0 + c];
    sum += __shfl_down(sum, 16);            // combine halves (wave32)
    const int row = row0 + l16;
    if (half == 0 && row < n) out[row] = sum + bc2[0];
}

// ============================================================
// host-side launcher
// ============================================================
extern "C" void kernel_launch(void* const* d_in, const int* in_sizes, int n_in,
                              void* d_out, int out_size, void* d_ws, size_t ws_size,
                              hipStream_t stream) {
    // input order: for i in 0..2: x_i, ei_i, W_i, att_s_i, att_d_i, bias_i ; Wc1, bc1, Wc2, bc2
    const float* Wc1 = (const float*)d_in[18];
    const float* bc1 = (const float*)d_in[19];
    const float* Wc2 = (const float*)d_in[20];
    const float* bc2 = (const float*)d_in[21];

    const int N = out_size;                       // output is [N,1]

    // workspace arena (floats):
    float* ws   = (float*)d_ws;
    float* XW   = ws;                             // N*128
    float* AGG  = XW   + (size_t)N * HC;          // N*128
    float* Hcat = AGG  + (size_t)N * HC;          // N*384
    float* asrc = Hcat + (size_t)N * FUSEDD;      // N*2
    float* adst = asrc + (size_t)N * HEADS;       // N*2
    float* mbuf = adst + (size_t)N * HEADS;       // N*2
    float* sbuf = mbuf + (size_t)N * HEADS;       // N*2

    const int B = 256;

    for (int i = 0; i < 3; ++i) {
        const float* x    = (const float*)d_in[6 * i + 0];
        const int*   ei   = (const int*)  d_in[6 * i + 1];
        const float* W    = (const float*)d_in[6 * i + 2];
        const float* ats  = (const float*)d_in[6 * i + 3];
        const float* atd  = (const float*)d_in[6 * i + 4];
        const float* bias = (const float*)d_in[6 * i + 5];
        const int din = in_sizes[6 * i + 2] / HC;  // W is [din, 128]
        const int E   = in_sizes[6 * i + 1] / 2;   // ei is [2, E]
        const int EN  = E + N;

        gat_gemm_xw<<<(N + 127) / 128, 256, 0, stream>>>(x, W, XW, N, din);
        gat_att_scores<<<(N * HEADS + B - 1) / B, B, 0, stream>>>(XW, ats, atd, asrc, adst, N);
        gat_init<<<(N * HC + B - 1) / B, B, 0, stream>>>(mbuf, sbuf, AGG, N);
        gat_edge_max<<<(EN + B - 1) / B, B, 0, stream>>>(ei, E, asrc, adst, mbuf, N);
        gat_edge_sum<<<(EN + B - 1) / B, B, 0, stream>>>(ei, E, asrc, adst, mbuf, sbuf, N);
        {
            long total = (long)EN * 32;
            long grid  = (total + B - 1) / B;
            gat_edge_aggregate<<<(unsigned)grid, B, 0, stream>>>(ei, E, asrc, adst, mbuf, sbuf, XW, AGG, N);
        }
        gat_elu_concat<<<(N * HC + B - 1) / B, B, 0, stream>>>(AGG, bias, Hcat, N, i);
    }

    fused_classifier<<<(N + 127) / 128, 256, 0, stream>>>(Hcat, Wc1, bc1, Wc2, bc2, (float*)d_out, N);
}